// ADC_VAR_55465207660710
// MI455X (gfx1250) — compile-verified
//
#include <hip/hip_runtime.h>
#include <hip/hip_bf16.h>
#include <stdint.h>

// ---------------------------------------------------------------------------
// ADC transfer-curve simulation, MI455X (gfx1250, wave32).
//  Pass 1: per-batch max   (reads 205.5 MB, RT policy -> stays hot in 192MB L2)
//  Pass 2: fused transform (NT reads mostly hitting L2, NT writes 205.5 MB)
//  Table (1KB) staged into LDS via the Tensor Data Mover (tensor_load_to_lds),
//  completion via s_wait_tensorcnt. Roofline floor ~26us @ 23.3 TB/s.
// ---------------------------------------------------------------------------

#define BATCHES        64
#define PER_BATCH      802816      // 256*56*56
#define V4_PER_BATCH   200704      // PER_BATCH / 4
#define MAX_BLOCKS_X   112         // 112*256 threads; 200704/(112*256) = 7 iters
#define TRF_BLOCKS_X   784         // 784*256 = 200704 float4 per batch exactly

typedef __attribute__((ext_vector_type(4))) unsigned int v4u;
typedef __attribute__((ext_vector_type(8))) int          v8i;
typedef __attribute__((ext_vector_type(4))) int          v4i;
typedef __attribute__((ext_vector_type(4))) float        v4f;   // NT-capable float4

// ---------------- RNG: counter hash + Box-Muller (seeded 42) ----------------
__device__ __forceinline__ unsigned long long mix64(unsigned long long z) {
    z ^= z >> 33; z *= 0xff51afd7ed558ccdULL;
    z ^= z >> 33; z *= 0xc4ceb9fe1a85ec53ULL;
    z ^= z >> 33;
    return z;
}

__device__ __forceinline__ void box_muller(unsigned long long h, float& n0, float& n1) {
    // two 24-bit uniforms in (0,1)
    float u0 = (float)(unsigned)(h >> 40)              * 0x1.0p-24f + 0x1.0p-25f;
    float u1 = (float)((unsigned)(h >> 8) & 0xFFFFFFu) * 0x1.0p-24f;
    float r  = __builtin_sqrtf(-2.0f * __logf(u0));
    float a  = 6.28318530718f * u1;
    n0 = r * __cosf(a);
    n1 = r * __sinf(a);
}

// ---------------- Pass 0: zero per-batch max slots (ws is poisoned) ---------
__global__ void adc_init_kernel(unsigned* __restrict__ bmax) {
    if (threadIdx.x < BATCHES) bmax[threadIdx.x] = 0u;
}

// ---------------- Pass 1: per-batch max (wave32 reduction) ------------------
__global__ void adc_batch_max_kernel(const v4f* __restrict__ x4,
                                     unsigned* __restrict__ bmax) {
    __shared__ float warp_part[8];
    const int b = blockIdx.y;
    const v4f* __restrict__ p = x4 + (long long)b * V4_PER_BATCH;

    float m = 0.0f;
    for (int i = blockIdx.x * blockDim.x + threadIdx.x; i < V4_PER_BATCH;
         i += gridDim.x * blockDim.x) {
        v4f v = p[i];                          // global_load_b128, RT (keep in L2)
        m = fmaxf(m, fmaxf(fmaxf(v.x, v.y), fmaxf(v.z, v.w)));
    }
    // wave32 tree (NOT 64: gfx1250 is wave32-only)
    #pragma unroll
    for (int off = 16; off > 0; off >>= 1)
        m = fmaxf(m, __shfl_xor(m, off, 32));

    const int lane = threadIdx.x & 31;
    const int wave = threadIdx.x >> 5;
    if (lane == 0) warp_part[wave] = m;
    __syncthreads();
    if (wave == 0) {
        m = (lane < 8) ? warp_part[lane] : 0.0f;
        #pragma unroll
        for (int off = 4; off > 0; off >>= 1)
            m = fmaxf(m, __shfl_xor(m, off, 32));
        if (lane == 0)
            atomicMax(&bmax[b], __float_as_uint(m));   // inputs >= 0: uint order == float order
    }
}

// ---------------- Pass 2: fused ADC transform -------------------------------
__global__ void adc_transform_kernel(const v4f* __restrict__ x4,
                                     const float* __restrict__ table_y,
                                     const unsigned* __restrict__ bmax,
                                     v4f* __restrict__ out4) {
    __shared__ float lds_table[256];

    // --- Stage the 256-entry transfer curve into LDS with the TDM -----------
    if (threadIdx.x < 32) {   // one wave issues the descriptor (EXEC ignored by TDM)
        unsigned long long ga = (unsigned long long)(uintptr_t)table_y;
        unsigned lds_off = (unsigned)(uintptr_t)(void*)lds_table; // low 32b = LDS offset
        // D# group0: count=1 | lds_addr | global_addr[56:0] | type=2
        v4u g0 = { 1u,
                   lds_off,
                   (unsigned)ga,
                   (unsigned)((ga >> 32) & 0x01FFFFFFu) | 0x80000000u };
        // D# group1: data_size=4B(code 2); tensor 256x1, tile 256x1, stride 256
        v8i g1 = { (int)0x00020000u,        // workgroup_mask=0, data_size=2
                   (int)(256u << 16),       // tensor_dim0 = 256 (bits 79:48, low half)
                   (int)(1u  << 16),        // tensor_dim1 = 1   (bits 111:80, low half)
                   (int)(256u << 16),       // tile_dim0   = 256 (bits 127:112)
                   (int)1,                  // tile_dim1   = 1   (bits 143:128)
                   (int)256,                // tensor_dim0_stride = 256 (bits 207:160)
                   0, 0 };
        v4i z4 = { 0, 0, 0, 0 };            // groups 2/3 unused (<=2-D tensor)
        v8i z8 = { 0, 0, 0, 0, 0, 0, 0, 0 };
        __builtin_amdgcn_tensor_load_to_lds(g0, g1, z4, z4, z8, 0);
        __builtin_amdgcn_s_wait_tensorcnt(0);
    }
    __syncthreads();

    // --- per-batch scale (uniform within block -> scalar regs) --------------
    const int b = blockIdx.y;
    const float mraw = __uint_as_float(bmax[b]);
    const float rr   = (mraw == 0.0f) ? 1.0f : mraw;
    const float s    = 255.0f / rr;
    const float back = rr * (1.0f / 256.0f);

    const long long g = (long long)b * V4_PER_BATCH
                      + (long long)blockIdx.x * blockDim.x + threadIdx.x;

    // NT load: this element is never touched again; don't churn L2 with it
    v4f v = __builtin_nontemporal_load(&x4[g]);

    // 4 gaussian gains for element indices 4g..4g+3 (flat order, "key 42")
    const unsigned long long e  = 4ull * (unsigned long long)g;
    const unsigned long long h0 = mix64(e        ^ 0x2A9E3779B97F4A7CULL);
    const unsigned long long h1 = mix64((e + 2u) ^ 0x2A9E3779B97F4A7CULL);
    float n0, n1, n2, n3;
    box_muller(h0, n0, n1);
    box_muller(h1, n2, n3);

    float xs[4] = { v.x, v.y, v.z, v.w };
    float ns[4] = { n0, n1, n2, n3 };
    float rs[4];
    #pragma unroll
    for (int k = 0; k < 4; ++k) {
        float r = fminf(fmaxf(xs[k] * s, 0.0f), 255.0f);   // ADC code
        int   idx = (int)r;  idx = idx > 254 ? 254 : idx;  // uniform-grid interp
        float frac = r - (float)idx;
        float t0 = lds_table[idx];                          // ds_load_b32 x2
        float t1 = lds_table[idx + 1];
        float y  = __builtin_fmaf(frac, t1 - t0, t0);
        float gn = __builtin_fmaf(0.05f, ns[k], 1.0f);      // 1 + 0.05*N(0,1)
        float q  = floorf(y * gn * (256.0f / 255.0f));      // 2^bits / scale
        rs[k] = q * back;
    }
    v4f o = { rs[0], rs[1], rs[2], rs[3] };
    __builtin_nontemporal_store(o, &out4[g]);               // NT: pure stream out
}

extern "C" void kernel_launch(void* const* d_in, const int* in_sizes, int n_in,
                              void* d_out, int out_size, void* d_ws, size_t ws_size,
                              hipStream_t stream) {
    const float* x       = (const float*)d_in[0];   // [64,256,56,56] f32
    // d_in[1] = table_x (linspace 0..255, uniform -> implicit), unused on device
    const float* table_y = (const float*)d_in[2];   // [256] f32
    float*       out     = (float*)d_out;
    unsigned*    bmax    = (unsigned*)d_ws;         // 64 per-batch max slots

    adc_init_kernel<<<1, 64, 0, stream>>>(bmax);

    dim3 gridMax(MAX_BLOCKS_X, BATCHES, 1);
    adc_batch_max_kernel<<<gridMax, 256, 0, stream>>>((const v4f*)x, bmax);

    dim3 gridTrf(TRF_BLOCKS_X, BATCHES, 1);
    adc_transform_kernel<<<gridTrf, 256, 0, stream>>>((const v4f*)x, table_y,
                                                      bmax, (v4f*)out);
    (void)in_sizes; (void)n_in; (void)out_size; (void)ws_size;
}